// MoEConnectionProcessor_36584531427888
// MI455X (gfx1250) — compile-verified
//
#include <hip/hip_runtime.h>
#include <math.h>

typedef float v2f __attribute__((ext_vector_type(2)));
typedef float v8f __attribute__((ext_vector_type(8)));
typedef int   v4i __attribute__((ext_vector_type(4)));

#define AS1 __attribute__((address_space(1)))
#define AS3 __attribute__((address_space(3)))

#if __has_builtin(__builtin_amdgcn_global_load_async_to_lds_b128) && \
    __has_builtin(__builtin_amdgcn_s_wait_asynccnt)
#define USE_ASYNC 1
#else
#define USE_ASYNC 0
#endif

#define WAVES 8
#define TPB (WAVES * 32)
#define CPW 16                 // cells per wave (== WMMA M)
#define CPB (WAVES * CPW)      // cells per block
#define KNB 26
#define DD  32

// fp32 WMMA: D(16x16) = A(16x4) * B(4x16) + C   -- full fp32 precision
__device__ __forceinline__ v8f wmma4(v2f a, v2f b, v8f c) {
  return __builtin_amdgcn_wmma_f32_16x16x4_f32(false, a, false, b, (short)0, c,
                                               false, false);
}

__device__ __forceinline__ void wait_async_le4() {
#if USE_ASYNC
  __builtin_amdgcn_s_wait_asynccnt(4);
#endif
}
__device__ __forceinline__ void wait_async_le0() {
#if USE_ASYNC
  __builtin_amdgcn_s_wait_asynccnt(0);
#endif
}

// C[16x32] = concat(U[16x32], V[16x32]) @ W[64x32] + bias[32]
// U,V,W,bias live in LDS (row-major). Result in WMMA C-layout:
// c0 -> cols ln (0..15), c1 -> cols ln+16; rows i + 8*half per VGPR i.
__device__ __forceinline__ void mm64(const float* U, const float* V,
                                     const float* W, const float* bias,
                                     int ln, int half, v8f& c0, v8f& c1) {
  v8f a0 = {0.f, 0.f, 0.f, 0.f, 0.f, 0.f, 0.f, 0.f};
  v8f a1 = {0.f, 0.f, 0.f, 0.f, 0.f, 0.f, 0.f, 0.f};
#pragma unroll
  for (int kb = 0; kb < 16; ++kb) {
    const float* src = (kb < 8) ? U : V;
    const int kl = (kb & 7) * 4 + 2 * half;   // k within the 16x32 source tile
    v2f a;
    a.x = src[ln * 32 + kl];
    a.y = src[ln * 32 + kl + 1];
    const int kg = kb * 4 + 2 * half;         // k within the 64-row weight
    v2f b0, b1;
    b0.x = W[kg * 32 + ln];        b0.y = W[(kg + 1) * 32 + ln];
    b1.x = W[kg * 32 + ln + 16];   b1.y = W[(kg + 1) * 32 + ln + 16];
    a0 = wmma4(a, b0, a0);
    a1 = wmma4(a, b1, a1);
  }
  a0 = a0 + bias[ln];
  a1 = a1 + bias[ln + 16];
  c0 = a0;
  c1 = a1;
}

__global__ __launch_bounds__(TPB) void moe_kernel(
    const float* __restrict__ x, const float* __restrict__ nbr,
    const int* __restrict__ tiers,
    const float* __restrict__ Wl, const float* __restrict__ bl,
    const float* __restrict__ Wm, const float* __restrict__ bm,
    const float* __restrict__ Wu, const float* __restrict__ bu,
    const float* __restrict__ Wc, const float* __restrict__ bc,
    const float* __restrict__ Wg1, const float* __restrict__ bg1,
    const float* __restrict__ Wg2, const float* __restrict__ bg2,
    float* __restrict__ out) {
  extern __shared__ float smem[];
  const int tid = threadIdx.x;
  const int lane = tid & 31;
  const int wave = tid >> 5;
  const int ln = lane & 15;
  const int half = lane >> 4;

  // ---------------- LDS layout ----------------
  float* sWl = smem;              // 2048
  float* sWm = sWl + 2048;        // 1024
  float* sWu = sWm + 1024;        // 2048
  float* sWc = sWu + 2048;        // 2048
  float* sWg1 = sWc + 2048;       // 2048
  float* sWg2 = sWg1 + 2048;      // 128 (32x4 padded)
  float* sb = sWg2 + 128;         // 192: bl|bm|bu|bc|bg1|bg2
  float* pw = sb + 192 + wave * 3072;  // per-wave region (3072 floats)
  float* xs = pw;                 // 512: current_state tile [16][32]
  float* vt = xs + 512;           // 512: shared V tile for concat matmuls
  float* ut = vt + 512;           // 512: CNF state tile
  float* sc0 = ut + 512;          // 512: nb tile buffer 0 (later h/out tile)
  float* sc1 = sc0 + 512;         // 512: nb tile buffer 1
  float* gts = sc1 + 512;         // 64 : gates [16][4]
  int* str = (int*)(gts + 64);    // 416 ints (+pad): tier ids [16][26]

  // ---------------- stage weights (block-wide) ----------------
  for (int i = tid; i < 2048; i += TPB) {
    sWl[i] = Wl[i]; sWu[i] = Wu[i]; sWc[i] = Wc[i]; sWg1[i] = Wg1[i];
  }
  for (int i = tid; i < 1024; i += TPB) sWm[i] = Wm[i];
  if (tid < 96) sWg2[(tid / 3) * 4 + (tid % 3)] = Wg2[tid];
  if (tid < 32) {
    sWg2[tid * 4 + 3] = 0.f;
    sb[tid] = bl[tid]; sb[32 + tid] = bm[tid]; sb[64 + tid] = bu[tid];
    sb[96 + tid] = bc[tid]; sb[128 + tid] = bg1[tid];
  }
  if (tid < 3) sb[160 + tid] = bg2[tid];
  __syncthreads();

  const int cellBase = blockIdx.x * CPB + wave * CPW;

  // neighbor-tile staging: async (ASYNCcnt) when available, else sync copy.
  auto stage = [&](float* dst, int kk) {
    for (int i = lane; i < 128; i += 32) {
      const int r = i >> 3, c = i & 7;
      const float* g = nbr + ((size_t)(cellBase + r) * KNB + kk) * DD + c * 4;
      float* l = dst + r * 32 + c * 4;
#if USE_ASYNC
      __builtin_amdgcn_global_load_async_to_lds_b128(
          (AS1 v4i*)(float*)const_cast<float*>(g), (AS3 v4i*)l, 0, 0);
#else
      *(float4*)l = *(const float4*)g;
#endif
    }
  };

  // ---------------- stage current_state + tiers ----------------
  for (int i = lane; i < 128; i += 32) {
    const int r = i >> 3, c = i & 7;
    ((float4*)xs)[r * 8 + c] =
        ((const float4*)(x + (size_t)(cellBase + r) * DD))[c];
  }
  for (int i = lane; i < CPW * KNB; i += 32)
    str[i] = tiers[(size_t)cellBase * KNB + i];

  // prefetch neighbor tile for kk = 0
  stage(sc0, 0);
  __builtin_amdgcn_wave_barrier();

  // preload W_msg B-fragments (loop-invariant over k)
  v2f Bm0[8], Bm1[8];
#pragma unroll
  for (int kb = 0; kb < 8; ++kb) {
    const int kg = kb * 4 + 2 * half;
    Bm0[kb].x = sWm[kg * 32 + ln];      Bm0[kb].y = sWm[(kg + 1) * 32 + ln];
    Bm1[kb].x = sWm[kg * 32 + ln + 16]; Bm1[kb].y = sWm[(kg + 1) * 32 + ln + 16];
  }
  const float bmsg0 = sb[32 + ln], bmsg1 = sb[32 + ln + 16];

  const v8f z8 = {0.f, 0.f, 0.f, 0.f, 0.f, 0.f, 0.f, 0.f};
  v8f sl0 = z8, sl1 = z8;  // local-tier neighbor sums   (C-layout)
  v8f sd0 = z8, sd1 = z8;  // distant-tier neighbor sums (C-layout)
  v8f sa0 = z8, sa1 = z8;  // plain neighbor sums        (C-layout)
  v8f af0 = z8, af1 = z8;  // functional msg agg         (C-layout)
  float cl[8] = {}, cf[8] = {}, cd[8] = {};

  // ---------------- neighbor loop (double-buffered) ----------------
  for (int kk = 0; kk < KNB; ++kk) {
    float* cur = (kk & 1) ? sc1 : sc0;
    float* nxt = (kk & 1) ? sc0 : sc1;
    if (kk < KNB - 1) {
      stage(nxt, kk + 1);   // prefetch next tile while computing this one
      wait_async_le4();     // drain this tile's 4 async ops, keep next 4 in flight
    } else {
      wait_async_le0();
    }
    __builtin_amdgcn_wave_barrier();

    float ffv[8];
#pragma unroll
    for (int i = 0; i < 8; ++i) {
      const int m = i + 8 * half;
      const int t = str[m * KNB + kk];
      const float fl = (t == 0) ? 1.f : 0.f;
      const float ff = (t == 1) ? 1.f : 0.f;
      const float fd = (t == 2) ? 1.f : 0.f;
      cl[i] += fl; cf[i] += ff; cd[i] += fd; ffv[i] = ff;
      const float n0 = cur[m * 32 + ln];
      const float n1 = cur[m * 32 + ln + 16];
      sl0[i] += fl * n0; sl1[i] += fl * n1;
      sd0[i] += fd * n0; sd1[i] += fd * n1;
      sa0[i] += n0;      sa1[i] += n1;
    }

    // msgs = nb_tile @ W_msg  (16 x v_wmma_f32_16x16x4_f32)
    v8f mc0 = z8, mc1 = z8;
#pragma unroll
    for (int kb = 0; kb < 8; ++kb) {
      const int k0 = kb * 4 + 2 * half;
      v2f a;
      a.x = cur[ln * 32 + k0];
      a.y = cur[ln * 32 + k0 + 1];
      mc0 = wmma4(a, Bm0[kb], mc0);
      mc1 = wmma4(a, Bm1[kb], mc1);
    }
#pragma unroll
    for (int i = 0; i < 8; ++i) {
      af0[i] += ffv[i] * tanhf(mc0[i] + bmsg0);
      af1[i] += ffv[i] * tanhf(mc1[i] + bmsg1);
    }
    __builtin_amdgcn_wave_barrier();
  }

  // ---------------- finalize masked means ----------------
#pragma unroll
  for (int i = 0; i < 8; ++i) {
    const float rl = 1.f / fmaxf(cl[i], 1.f);
    const float rf = 1.f / fmaxf(cf[i], 1.f);
    const float rd = 1.f / fmaxf(cd[i], 1.f);
    sl0[i] *= rl; sl1[i] *= rl;
    sd0[i] *= rd; sd1[i] *= rd;
    af0[i] *= rf; af1[i] *= rf;
    sa0[i] *= (1.f / (float)KNB); sa1[i] *= (1.f / (float)KNB);
  }

  // current_state in C-layout
  float xc0[8], xc1[8];
#pragma unroll
  for (int i = 0; i < 8; ++i) {
    const int m = i + 8 * half;
    xc0[i] = xs[m * 32 + ln];
    xc1[i] = xs[m * 32 + ln + 16];
  }

  auto dump = [&](float* t, const v8f& c0, const v8f& c1) {
#pragma unroll
    for (int i = 0; i < 8; ++i) {
      const int m = i + 8 * half;
      t[m * 32 + ln] = c0[i];
      t[m * 32 + ln + 16] = c1[i];
    }
  };

  // ---- local expert: tanh([x | loc_mean] @ W_local + b) ----
  dump(vt, sl0, sl1);
  __builtin_amdgcn_wave_barrier();
  v8f lo0, lo1;
  mm64(xs, vt, sWl, sb + 0, ln, half, lo0, lo1);
  float l0[8], l1[8];
#pragma unroll
  for (int i = 0; i < 8; ++i) { l0[i] = tanhf(lo0[i]); l1[i] = tanhf(lo1[i]); }
  __builtin_amdgcn_wave_barrier();

  // ---- functional expert: z = sigmoid([x | agg] @ W_upd + b) ----
  dump(vt, af0, af1);
  __builtin_amdgcn_wave_barrier();
  v8f zz0, zz1;
  mm64(xs, vt, sWu, sb + 64, ln, half, zz0, zz1);
  float fo0[8], fo1[8];
#pragma unroll
  for (int i = 0; i < 8; ++i) {
    const float za = 1.f / (1.f + expf(-zz0[i]));
    const float zb = 1.f / (1.f + expf(-zz1[i]));
    fo0[i] = (1.f - za) * xc0[i] + za * tanhf(af0[i]);
    fo1[i] = (1.f - zb) * xc1[i] + zb * tanhf(af1[i]);
  }
  __builtin_amdgcn_wave_barrier();

  // ---- gating: h = relu([x | mean_nb] @ W_g1 + b), gates = softmax(h@W_g2+b) ----
  dump(vt, sa0, sa1);
  __builtin_amdgcn_wave_barrier();
  v8f h0, h1;
  mm64(xs, vt, sWg1, sb + 128, ln, half, h0, h1);
#pragma unroll
  for (int i = 0; i < 8; ++i) { h0[i] = fmaxf(h0[i], 0.f); h1[i] = fmaxf(h1[i], 0.f); }
  dump(sc0, h0, h1);
  __builtin_amdgcn_wave_barrier();
  {
    float ga = sb[160], gb = sb[161], gc = sb[162];
#pragma unroll
    for (int j = 0; j < 32; ++j) {
      const float hv = sc0[ln * 32 + j];
      ga += hv * sWg2[j * 4 + 0];
      gb += hv * sWg2[j * 4 + 1];
      gc += hv * sWg2[j * 4 + 2];
    }
    const float mx = fmaxf(ga, fmaxf(gb, gc));
    const float ea = expf(ga - mx), eb = expf(gb - mx), ec = expf(gc - mx);
    const float inv = 1.f / (ea + eb + ec);
    gts[ln * 4 + 0] = ea * inv;
    gts[ln * 4 + 1] = eb * inv;
    gts[ln * 4 + 2] = ec * inv;
    gts[ln * 4 + 3] = 0.f;
  }
  __builtin_amdgcn_wave_barrier();

  // ---- distant expert: 3-step Euler CNF ----
  dump(vt, sd0, sd1);
  __builtin_amdgcn_wave_barrier();
  float xq0[8], xq1[8];
#pragma unroll
  for (int i = 0; i < 8; ++i) { xq0[i] = xc0[i]; xq1[i] = xc1[i]; }
  const float dt = 1.f / 3.f;
  for (int s = 0; s < 3; ++s) {
#pragma unroll
    for (int i = 0; i < 8; ++i) {
      const int m = i + 8 * half;
      ut[m * 32 + ln] = xq0[i];
      ut[m * 32 + ln + 16] = xq1[i];
    }
    __builtin_amdgcn_wave_barrier();
    v8f v0, v1;
    mm64(ut, vt, sWc, sb + 96, ln, half, v0, v1);
#pragma unroll
    for (int i = 0; i < 8; ++i) {
      xq0[i] += dt * tanhf(v0[i]);
      xq1[i] += dt * tanhf(v1[i]);
    }
    __builtin_amdgcn_wave_barrier();
  }

  // ---- combine and store ----
#pragma unroll
  for (int i = 0; i < 8; ++i) {
    const int m = i + 8 * half;
    const float ga = gts[m * 4 + 0], gb = gts[m * 4 + 1], gc = gts[m * 4 + 2];
    sc0[m * 32 + ln]      = ga * l0[i] + gb * fo0[i] + gc * xq0[i];
    sc0[m * 32 + ln + 16] = ga * l1[i] + gb * fo1[i] + gc * xq1[i];
  }
  __builtin_amdgcn_wave_barrier();
  for (int i = lane; i < 128; i += 32) {
    const int r = i >> 3, c = i & 7;
    ((float4*)(out + (size_t)(cellBase + r) * DD))[c] = ((float4*)sc0)[r * 8 + c];
  }
}

extern "C" void kernel_launch(void* const* d_in, const int* in_sizes, int n_in,
                              void* d_out, int out_size, void* d_ws,
                              size_t ws_size, hipStream_t stream) {
  (void)n_in; (void)out_size; (void)d_ws; (void)ws_size;
  const float* x = (const float*)d_in[0];
  const float* nbr = (const float*)d_in[1];
  const int* tid = (const int*)d_in[2];
  const float* Wl = (const float*)d_in[3];
  const float* bl = (const float*)d_in[4];
  const float* Wm = (const float*)d_in[5];
  const float* bm = (const float*)d_in[6];
  const float* Wu = (const float*)d_in[7];
  const float* bu = (const float*)d_in[8];
  const float* Wc = (const float*)d_in[9];
  const float* bc = (const float*)d_in[10];
  const float* Wg1 = (const float*)d_in[11];
  const float* bg1 = (const float*)d_in[12];
  const float* Wg2 = (const float*)d_in[13];
  const float* bg2 = (const float*)d_in[14];

  const int B = in_sizes[0] / DD;       // 262144
  const int blocks = B / CPB;           // 2048
  const size_t shmem = (9536 + WAVES * 3072) * sizeof(float);  // 136,448 B

  (void)hipFuncSetAttribute((const void*)moe_kernel,
                            hipFuncAttributeMaxDynamicSharedMemorySize,
                            (int)shmem);
  moe_kernel<<<blocks, TPB, shmem, stream>>>(x, nbr, tid, Wl, bl, Wm, bm, Wu,
                                             bu, Wc, bc, Wg1, bg1, Wg2, bg2,
                                             (float*)d_out);
}